// EntropyController_10299331575865
// MI455X (gfx1250) — compile-verified
//
#include <hip/hip_runtime.h>
#include <hip/hip_bf16.h>

// EntropyController for MI455X (gfx1250, wave32).
//
// Only controlled_features and control_signal are live outputs; the MLP
// estimator branch is dead code in the reference and is skipped.
//
// Heavy work: Gram matrix X@X^T (N=8192, D=256, f32) -> per-row kernel sums,
// via V_WMMA_F32_16X16X4_F32 in a single pass using
//   H_i = log(S_i) + T_i/S_i,  S = sum_j exp(-e_ij), T = sum_j e_ij*exp(-e_ij),
//   e_ij = max(||xi||^2+||xj||^2-2 xi.xj, 0) / (2 t^2)
// (the reference's +1e-6 inside the log contributes at most N*eps=0.008 to H,
//  i.e. <4e-5 after the sigmoid; neglected).
//
// Double-buffered B tiles in LDS (one barrier / tile, fetch latency hidden
// under the WMMA chain), two interleaved WMMA accumulator chains, and
// GLOBAL_LOAD_ASYNC_TO_LDS_B128 staging (ASYNCcnt-tracked).

typedef __attribute__((ext_vector_type(2))) float v2f;
typedef __attribute__((ext_vector_type(8))) float v8f;

#define NROWS 8192           // B*S = 4*2048
#define D     256            // feature dim
#define LDB   260            // LDS pitch (floats): 260 % 64 == 4 -> conflict-free b64 reads
#define BUFSZ (16 * LDB)     // one B tile buffer (floats)

#if defined(__has_builtin)
#if __has_builtin(__builtin_amdgcn_global_load_async_to_lds_b128) && \
    __has_builtin(__builtin_amdgcn_s_wait_asynccnt)
#define USE_ASYNC_LDS 1
#endif
#endif

#if defined(USE_ASYNC_LDS)
// Builtin signature (from hipcc diagnostic): (int4 AS1*, int4 AS3*, Ii, Ii)
typedef __attribute__((__vector_size__(4 * sizeof(int)))) int v4i_t;
typedef __attribute__((address_space(1))) v4i_t g_v4i;
typedef __attribute__((address_space(3))) v4i_t l_v4i;
__device__ __forceinline__ void async_copy16(const float* gsrc, float* ldst) {
  // Flat LDS addresses carry the LDS byte offset in the low 32 bits (ISA 10.2),
  // and flat global addresses equal their AS(1) addresses.
  __builtin_amdgcn_global_load_async_to_lds_b128(
      (g_v4i*)(uintptr_t)gsrc, (l_v4i*)(uint32_t)(uintptr_t)ldst,
      /*offset=*/0, /*cpol=*/0);
}
#endif

// ---------------------------------------------------------------- row norms
__global__ void sq_kernel(const float* __restrict__ x, float* __restrict__ sq) {
  int row = blockIdx.x * blockDim.x + threadIdx.x;
  if (row >= NROWS) return;
  const float4* p = (const float4*)(x + (size_t)row * D);
  float s = 0.0f;
#pragma unroll
  for (int i = 0; i < D / 4; ++i) {
    float4 v = p[i];
    s += v.x * v.x + v.y * v.y + v.z * v.z + v.w * v.w;
  }
  sq[row] = s;
}

// ------------------------------------------------- Gram + entropy sums (WMMA)
// One wave = one 16-row strip of the NxN matrix. 8 waves/block share B tiles
// double-buffered in LDS. A strip lives in registers as 64 v2f WMMA fragments.
__global__ void __launch_bounds__(256)
gram_entropy_kernel(const float* __restrict__ x,
                    const float* __restrict__ sq,
                    const float* __restrict__ temperature,
                    float* __restrict__ Ssum,
                    float* __restrict__ Tsum) {
  __shared__ float Bs[2 * BUFSZ];

  const int tid  = threadIdx.x;
  const int lane = tid & 31;
  const int wave = tid >> 5;                  // 0..7
  const int half = lane >> 4;                 // 0: K lo pair, 1: K hi pair
  const int l15  = lane & 15;
  const int i0   = (blockIdx.x * 8 + wave) * 16;

  // Cooperative staging geometry: thread -> (row, 16-float column chunk).
  const int sr = tid >> 4;                    // 0..15
  const int sc = (tid & 15) * 16;             // 0..240

  const float t      = temperature[0];
  const float inv2t2 = 1.0f / (2.0f * t * t);

  // A-matrix 16x4 f32 fragment layout (ISA 7.12.2):
  //   VGPR0: lanes0-15 K=k0+0, lanes16-31 K=k0+2 ; VGPR1: K=k0+1 / K=k0+3
  // => per lane a v2f at X[i0 + (lane&15), k0 + 2*half].
  v2f aFrag[64];
  {
    const float* arow = x + (size_t)(i0 + l15) * D + half * 2;
#pragma unroll
    for (int kk = 0; kk < 64; ++kk)
      aFrag[kk] = *(const v2f*)(arow + kk * 4);
  }

  // per-lane ||x_i||^2 for the 8 rows this lane's accumulator covers
  float sqi[8];
#pragma unroll
  for (int v = 0; v < 8; ++v) sqi[v] = sq[i0 + half * 8 + v];

  float Sp[8], Tp[8];
#pragma unroll
  for (int v = 0; v < 8; ++v) { Sp[v] = 0.0f; Tp[v] = 0.0f; }

  // ---- prologue: stage tile j0 = 0 into buffer 0
  {
    const float* src = x + (size_t)sr * D + sc;
    float* dst = &Bs[sr * LDB + sc];
#if defined(USE_ASYNC_LDS)
    async_copy16(src + 0,  dst + 0);
    async_copy16(src + 4,  dst + 4);
    async_copy16(src + 8,  dst + 8);
    async_copy16(src + 12, dst + 12);
    __builtin_amdgcn_s_wait_asynccnt(0);
#else
    const float4* s4 = (const float4*)src;
    float4 v0 = s4[0], v1 = s4[1], v2 = s4[2], v3 = s4[3];
    ((float4*)dst)[0] = v0;
    *(float4*)(dst + 4)  = v1;
    *(float4*)(dst + 8)  = v2;
    *(float4*)(dst + 12) = v3;
#endif
  }
  __syncthreads();

  int p = 0;
  for (int j0 = 0; j0 < NROWS; j0 += 16) {
    const int jn = j0 + 16;
    const float* bcur = &Bs[p * BUFSZ];
    float* bnxt = &Bs[(p ^ 1) * BUFSZ];

    // ---- prefetch next B tile while this tile is consumed by WMMA
#if defined(USE_ASYNC_LDS)
    if (jn < NROWS) {
      const float* src = x + (size_t)(jn + sr) * D + sc;
      float* dst = bnxt + sr * LDB + sc;
      async_copy16(src + 0,  dst + 0);
      async_copy16(src + 4,  dst + 4);
      async_copy16(src + 8,  dst + 8);
      async_copy16(src + 12, dst + 12);
    }
#else
    float4 r0, r1, r2, r3;
    if (jn < NROWS) {
      const float4* s4 = (const float4*)(x + (size_t)(jn + sr) * D + sc);
      r0 = s4[0]; r1 = s4[1]; r2 = s4[2]; r3 = s4[3];
    }
#endif
    const float sqj = sq[j0 + l15];   // overlaps the WMMA chain

    // ---- 16x16 tile of X X^T over K=256: two interleaved accumulator chains
    v8f acc0 = {0.f, 0.f, 0.f, 0.f, 0.f, 0.f, 0.f, 0.f};
    v8f acc1 = {0.f, 0.f, 0.f, 0.f, 0.f, 0.f, 0.f, 0.f};
    const float* brow = bcur + l15 * LDB + half * 2;
#pragma unroll
    for (int kk = 0; kk < 64; kk += 2) {
      v2f b0 = *(const v2f*)(brow + kk * 4);
      v2f b1 = *(const v2f*)(brow + kk * 4 + 4);
      acc0 = __builtin_amdgcn_wmma_f32_16x16x4_f32(
          false, aFrag[kk], false, b0, (short)0, acc0, false, false);
      acc1 = __builtin_amdgcn_wmma_f32_16x16x4_f32(
          false, aFrag[kk + 1], false, b1, (short)0, acc1, false, false);
    }

    // acc[v] = C[M,N] with M = v + 8*half, N = lane&15 (ISA C/D layout)
#pragma unroll
    for (int v = 0; v < 8; ++v) {
      float dot = acc0[v] + acc1[v];
      float d2 = sqi[v] + sqj - 2.0f * dot;
      d2 = fmaxf(d2, 0.0f);
      float e = d2 * inv2t2;
      float w = __expf(-e);
      Sp[v] += w;
      Tp[v] += e * w;
    }

    // ---- publish next buffer, single barrier per tile
#if defined(USE_ASYNC_LDS)
    __builtin_amdgcn_s_wait_asynccnt(0);
#else
    if (jn < NROWS) {
      float* dst = bnxt + sr * LDB + sc;
      ((float4*)dst)[0] = r0;
      *(float4*)(dst + 4)  = r1;
      *(float4*)(dst + 8)  = r2;
      *(float4*)(dst + 12) = r3;
    }
#endif
    __syncthreads();
    p ^= 1;
  }

  // Reduce the 16 lanes (same half) that share each row M.
#pragma unroll
  for (int v = 0; v < 8; ++v) {
#pragma unroll
    for (int m = 1; m < 16; m <<= 1) {
      Sp[v] += __shfl_xor(Sp[v], m, 32);
      Tp[v] += __shfl_xor(Tp[v], m, 32);
    }
  }
  if (l15 == 0) {
#pragma unroll
    for (int v = 0; v < 8; ++v) {
      int row = i0 + half * 8 + v;
      Ssum[row] = Sp[v];
      Tsum[row] = Tp[v];
    }
  }
}

// ----------------------------------------------------- control signal per row
__global__ void control_kernel(const float* __restrict__ Ssum,
                               const float* __restrict__ Tsum,
                               const float* __restrict__ target,
                               const float* __restrict__ temperature,
                               float* __restrict__ control) {
  int row = blockIdx.x * blockDim.x + threadIdx.x;
  if (row >= NROWS) return;
  float S = Ssum[row];
  float T = Tsum[row];
  float H = __logf(S) + T / S;                 // entropy (natural log)
  float z = -(H - target[0]) / temperature[0];
  control[row] = 1.0f / (1.0f + __expf(-z));   // sigmoid
}

// --------------------------------------------------------- scale the features
__global__ void scale_kernel(const float* __restrict__ x,
                             const float* __restrict__ control,
                             float* __restrict__ out) {
  int idx = blockIdx.x * blockDim.x + threadIdx.x;   // one float4 per thread
  int row = idx >> 6;                                 // (idx*4)/256
  float c = control[row];
  float4 v = ((const float4*)x)[idx];
  v.x *= c; v.y *= c; v.z *= c; v.w *= c;
  ((float4*)out)[idx] = v;
}

// ------------------------------------------------------------------- launcher
extern "C" void kernel_launch(void* const* d_in, const int* in_sizes, int n_in,
                              void* d_out, int out_size, void* d_ws, size_t ws_size,
                              hipStream_t stream) {
  (void)in_sizes; (void)n_in; (void)out_size; (void)ws_size;
  const float* features    = (const float*)d_in[0];
  const float* target      = (const float*)d_in[7];
  const float* temperature = (const float*)d_in[8];

  float* out_features = (float*)d_out;                       // 4*2048*256
  float* control      = (float*)d_out + (size_t)NROWS * D;   // 4*2048

  float* sq   = (float*)d_ws;        // NROWS
  float* Ssum = sq + NROWS;          // NROWS
  float* Tsum = Ssum + NROWS;        // NROWS

  sq_kernel<<<NROWS / 256, 256, 0, stream>>>(features, sq);
  gram_entropy_kernel<<<NROWS / (16 * 8), 256, 0, stream>>>(
      features, sq, temperature, Ssum, Tsum);
  control_kernel<<<NROWS / 256, 256, 0, stream>>>(Ssum, Tsum, target,
                                                  temperature, control);
  scale_kernel<<<(NROWS * D / 4) / 256, 256, 0, stream>>>(features, control,
                                                          out_features);
}